// TripletLoss_86096914415968
// MI455X (gfx1250) — compile-verified
//
#include <hip/hip_runtime.h>
#include <hip/hip_bf16.h>
#include <math.h>

// fp32 WMMA fragment types (16x16x4_f32): A = 16x4 -> 2 VGPRs/lane,
// B = 4x16 -> 2 VGPRs/lane, C/D = 16x16 f32 -> 8 VGPRs/lane.
typedef float v2f __attribute__((ext_vector_type(2)));
typedef float v8f __attribute__((ext_vector_type(8)));

#define DIM     128
#define KSTEPS  (DIM / 4)     // 32 wmma (K=4 each) per 16x16 tile
#define MARGIN  0.1f

// Each block: 128 rows (8 waves x 16-row panels). Sweep 1: masked row maxes.
// Sweep 2: thresholded loss sums + last-row diagnostics.
__global__ __launch_bounds__(256)
void triplet_main(const float* __restrict__ X, const int* __restrict__ T,
                  float* __restrict__ partial, float* __restrict__ out, int N)
{
    const int lane  = threadIdx.x & 31;
    const int wave  = threadIdx.x >> 5;      // 0..7
    const int lrow  = lane & 15;             // row (A/B frag) / col (C/D frag)
    const int khalf = lane >> 4;             // which K-pair of the K=4 slice
    const int Rtile = blockIdx.x * 128 + wave * 16;

    // ---- A fragment: resident for the whole kernel (64 VGPRs) ----
    v2f a[KSTEPS];
    const float* arow = X + (size_t)(Rtile + lrow) * DIM + khalf * 2;
#pragma unroll
    for (int ks = 0; ks < KSTEPS; ++ks)
        a[ks] = *reinterpret_cast<const v2f*>(arow + ks * 4);

    // Row labels for the 8 C/D rows this lane sees (M = 8*khalf + v)
    int trow[8];
#pragma unroll
    for (int v = 0; v < 8; ++v)
        trow[v] = T[Rtile + khalf * 8 + v];

    const int ntiles = N / 16;

    // ================= Sweep 1: masked row maxes =================
    float maxpos[8], maxneg[8];
#pragma unroll
    for (int v = 0; v < 8; ++v) { maxpos[v] = -INFINITY; maxneg[v] = -INFINITY; }

    for (int jt = 0; jt < ntiles; ++jt) {
        const float* brow = X + (size_t)(jt * 16 + lrow) * DIM + khalf * 2;
        const int tcol = T[jt * 16 + lrow];
        v8f acc = {};
#pragma unroll
        for (int ks = 0; ks < KSTEPS; ++ks) {
            v2f b = *reinterpret_cast<const v2f*>(brow + ks * 4);
            acc = __builtin_amdgcn_wmma_f32_16x16x4_f32(
                false, a[ks], false, b, (short)0, acc, false, false);
        }
        const int gc = jt * 16 + lrow;
#pragma unroll
        for (int v = 0; v < 8; ++v) {
            const int gr = Rtile + khalf * 8 + v;
            const float s = acc[v];
            if (trow[v] == tcol) {
                if (gr != gc) maxpos[v] = fmaxf(maxpos[v], s);
            } else {
                maxneg[v] = fmaxf(maxneg[v], s);
            }
        }
    }
    // Butterfly reduce over the 16-lane column group (stays within each half)
#pragma unroll
    for (int v = 0; v < 8; ++v) {
#pragma unroll
        for (int m = 1; m <= 8; m <<= 1) {
            maxpos[v] = fmaxf(maxpos[v], __shfl_xor(maxpos[v], m, 32));
            maxneg[v] = fmaxf(maxneg[v], __shfl_xor(maxneg[v], m, 32));
        }
    }
    float thr_p[8], thr_n[8];
    bool  haspos[8];
#pragma unroll
    for (int v = 0; v < 8; ++v) {
        thr_p[v]  = maxneg[v] + MARGIN;                   // pos selected if s <  thr_p
        thr_n[v]  = fmaxf(0.6f, maxpos[v]) - MARGIN;      // neg selected if s >  thr_n
        haspos[v] = (maxpos[v] > -INFINITY);
    }

    // ================= Sweep 2: thresholded sums =================
    float loss[8];
#pragma unroll
    for (int v = 0; v < 8; ++v) loss[v] = 0.0f;
    float psum = 0.f, pcnt = 0.f, nsum = 0.f, ncnt = 0.f;  // last-row diagnostics

    for (int jt = 0; jt < ntiles; ++jt) {
        const float* brow = X + (size_t)(jt * 16 + lrow) * DIM + khalf * 2;
        const int tcol = T[jt * 16 + lrow];
        v8f acc = {};
#pragma unroll
        for (int ks = 0; ks < KSTEPS; ++ks) {
            v2f b = *reinterpret_cast<const v2f*>(brow + ks * 4);
            acc = __builtin_amdgcn_wmma_f32_16x16x4_f32(
                false, a[ks], false, b, (short)0, acc, false, false);
        }
        const int gc = jt * 16 + lrow;
#pragma unroll
        for (int v = 0; v < 8; ++v) {
            const int gr = Rtile + khalf * 8 + v;
            const float s = acc[v];
            if (trow[v] == tcol) {
                if (gr != gc) {
                    if (s < thr_p[v]) loss[v] += 1.0f - s;
                    if (gr == N - 1) { psum += s; pcnt += 1.0f; }
                }
            } else {
                if (s > thr_n[v]) loss[v] += s;
                if (gr == N - 1) { nsum += s; ncnt += 1.0f; }
            }
        }
    }

    // Per-wave loss total (has_pos gating is row-uniform across the group)
    float wsum = 0.0f;
#pragma unroll
    for (int v = 0; v < 8; ++v)
        if (haspos[v]) wsum += loss[v];
#pragma unroll
    for (int m = 1; m <= 16; m <<= 1)
        wsum += __shfl_xor(wsum, m, 32);

    __shared__ float bsum[8];
    if (lane == 0) bsum[wave] = wsum;
    __syncthreads();
    if (threadIdx.x == 0) {
        float t = 0.f;
#pragma unroll
        for (int w = 0; w < 8; ++w) t += bsum[w];
        partial[blockIdx.x] = t;
    }

    // Last-row diagnostics: reduce within the khalf==1 group, lane 16 writes
    if (Rtile + 15 == N - 1) {
#pragma unroll
        for (int m = 1; m <= 8; m <<= 1) {
            psum += __shfl_xor(psum, m, 32);
            pcnt += __shfl_xor(pcnt, m, 32);
            nsum += __shfl_xor(nsum, m, 32);
            ncnt += __shfl_xor(ncnt, m, 32);
        }
        if (lane == 16) {
            out[4] = psum / fmaxf(pcnt, 1.0f);
            out[5] = nsum / fmaxf(ncnt, 1.0f);
        }
    }
}

// Deterministic fixed-order reduction of per-block partials (no float atomics).
__global__ void triplet_final(const float* __restrict__ partial,
                              float* __restrict__ out, int nblocks, float invN)
{
    if (threadIdx.x == 0 && blockIdx.x == 0) {
        float t = 0.f;
        for (int i = 0; i < nblocks; ++i) t += partial[i];
        out[0] = t * invN;
        out[1] = 0.f; out[2] = 0.f; out[3] = 0.f;
    }
}

extern "C" void kernel_launch(void* const* d_in, const int* in_sizes, int n_in,
                              void* d_out, int out_size, void* d_ws, size_t ws_size,
                              hipStream_t stream)
{
    const float* X = (const float*)d_in[0];   // [N, 128] fp32, L2-normalized rows
    const int*   T = (const int*)d_in[1];     // [N] labels
    float* out     = (float*)d_out;           // 6 scalars
    float* partial = (float*)d_ws;            // one float per block

    const int N = in_sizes[1];                // 8192
    const int nblocks = N / 128;              // 64

    triplet_main<<<nblocks, 256, 0, stream>>>(X, T, partial, out, N);
    triplet_final<<<1, 1, 0, stream>>>(partial, out, nblocks, 1.0f / (float)N);
}